// TemporalLogicMachineDP2D_72980084294238
// MI455X (gfx1250) — compile-verified
//
#include <hip/hip_runtime.h>
#include <math.h>

// ---------------------------------------------------------------------------
// TemporalLogicMachineDP2D on MI455X (gfx1250)
// Math: max_k sigmoid(x_k) == sigmoid(max_k x_k)  (sigmoid monotone)
//  => dp_step = tropical (max-plus) matmul over k, sigmoid in epilogue.
// Dense GEMMs (rows=4*97*97, K={32,48}, N=16) use V_WMMA_F32_16X16X4_F32
// (exact fp32 accumulate).
// ---------------------------------------------------------------------------

typedef __attribute__((ext_vector_type(2))) float v2f;
typedef __attribute__((ext_vector_type(8))) float v8f;

#define TT    96
#define TP    97
#define NB    4
#define MAT   (TP * TP)       // 9409
#define NROWS (NB * MAT)      // 37636
#define RELC  32
#define NEGF  (-3.0e38f)
#define POSF  ( 3.0e38f)

// ---------------- interval pooling (max & min), writes into padded r --------
// r_pad layout: [b][i][jj][c], (B,97,97,32); r_pad[b,i,j+1, d]=max, [d+16]=min
// pass1: per (b,j,d): reverse scan over i (i = j..0): running max/min of a[.,i,j,d]
__global__ void pool_pass1(const float* __restrict__ a, float* __restrict__ rp) {
  int idx = blockIdx.x * blockDim.x + threadIdx.x;
  if (idx >= NB * TT * 16) return;
  int d = idx & 15;
  int j = (idx >> 4) % TT;
  int b = idx / (TT * 16);
  float rmax = NEGF, rmin = POSF;
  for (int i = j; i >= 0; --i) {
    float v = a[(((b * TT) + i) * TT + j) * 16 + d];
    rmax = fmaxf(rmax, v);
    rmin = fminf(rmin, v);
    int base = (((b * TP) + i) * TP + (j + 1)) * RELC;
    rp[base + d]      = rmax;
    rp[base + 16 + d] = rmin;
  }
}

// pass2: per (b,i,d): forward scan over j (j = i..95), in place
__global__ void pool_pass2(float* __restrict__ rp) {
  int idx = blockIdx.x * blockDim.x + threadIdx.x;
  if (idx >= NB * TT * 16) return;
  int d = idx & 15;
  int i = (idx >> 4) % TT;
  int b = idx / (TT * 16);
  float rmax = NEGF, rmin = POSF;
  for (int j = i; j < TT; ++j) {
    int base = (((b * TP) + i) * TP + (j + 1)) * RELC;
    rmax = fmaxf(rmax, rp[base + d]);      rp[base + d]      = rmax;
    rmin = fminf(rmin, rp[base + 16 + d]); rp[base + 16 + d] = rmin;
  }
}

// ---------------- WMMA fp32 GEMM: C[NROWS x 16] = [A1|A2] @ W ---------------
// One wave per 16-row tile; K in chunks of 4 via V_WMMA_F32_16X16X4_F32.
// A frag (16x4 f32): lane L holds A[L&15][k0], A[L&15][k0+1], k0 = (L>=16)?2:0
// B frag (4x16 f32): lane L holds W[k0][L&15], W[k0+1][L&15]   (half-split K)
// C frag (16x16 f32): VGPR e holds M=e (lanes 0-15) / M=e+8 (lanes 16-31)
// mode 0: store transposed per (b,h): Out[(b*16+h)*9409 + i*97 + k]
// mode 2: sigmoid(c + bias), slice [:, :96, 1:, :] -> Out (B,96,96,16)
__global__ void wmma_gemm16(const float* __restrict__ A1, int K1,
                            const float* __restrict__ A2, int K2,
                            const float* __restrict__ W,
                            float* __restrict__ Out, int mode,
                            const float* __restrict__ bias) {
  int wave = threadIdx.x >> 5;
  int lane = threadIdx.x & 31;
  int tile = blockIdx.x * (blockDim.x >> 5) + wave;
  int rowbase = tile * 16;
  if (rowbase >= NROWS) return;            // wave-uniform exit

  int Ktot  = K1 + K2;
  int half  = lane >> 4;                   // 0 or 1
  int lan15 = lane & 15;
  int koff  = half * 2;
  int arow  = rowbase + lan15;
  if (arow >= NROWS) arow = NROWS - 1;     // clamp loads; stores are guarded

  v8f c = {0.f, 0.f, 0.f, 0.f, 0.f, 0.f, 0.f, 0.f};
  for (int kb = 0; kb < Ktot; kb += 4) {
    int k0 = kb + koff;
    v2f afrag, bfrag;
    if (k0 < K1) {
      const float* p = A1 + (size_t)arow * K1 + k0;
      afrag.x = p[0]; afrag.y = p[1];
    } else {
      const float* p = A2 + (size_t)arow * K2 + (k0 - K1);
      afrag.x = p[0]; afrag.y = p[1];
    }
    bfrag.x = W[k0 * 16 + lan15];
    bfrag.y = W[(k0 + 1) * 16 + lan15];
    c = __builtin_amdgcn_wmma_f32_16x16x4_f32(false, afrag, false, bfrag,
                                              (short)0, c, false, false);
  }

  int n = lan15;
  if (mode == 0) {
#pragma unroll
    for (int e = 0; e < 8; ++e) {
      int r = rowbase + e + half * 8;
      if (r < NROWS) {
        int b   = r / MAT;
        int rem = r - b * MAT;             // i*97 + k
        Out[(size_t)(b * 16 + n) * MAT + rem] = c[e];
      }
    }
  } else {
    float bb = bias[n];
#pragma unroll
    for (int e = 0; e < 8; ++e) {
      int r = rowbase + e + half * 8;
      if (r < NROWS) {
        int b   = r / MAT;
        int rem = r - b * MAT;
        int i   = rem / TP;
        int jj  = rem - i * TP;
        if (i < TT && jj >= 1) {
          float v = 1.0f / (1.0f + __expf(-(c[e] + bb)));
          Out[((size_t)((b * TT + i) * TT) + (jj - 1)) * 16 + n] = v;
        }
      }
    }
  }
}

// ---------------- tropical (max-plus) matmul per (b,h) ----------------------
// FT: [(b*16+h)][i][k] (97x97), RT: [(b*16+h)][k][j] (97x97)
// OUT[b,i,j,h] = sigmoid( max_k (FT[i,k] + RT[k,j]) + bias[h] ), row-major
__global__ void tropical_maxplus(const float* __restrict__ FT,
                                 const float* __restrict__ RT,
                                 const float* __restrict__ bias,
                                 float* __restrict__ OUT) {
  int bh = blockIdx.z;                  // b*16 + h
  const float* A = FT + (size_t)bh * MAT;
  const float* Bm = RT + (size_t)bh * MAT;
  int i0 = blockIdx.y * 16, j0 = blockIdx.x * 16;
  int ty = threadIdx.y, tx = threadIdx.x;

  __shared__ float As[16][17];
  __shared__ float Bs[16][17];

  float acc = -1.0e30f;
  for (int kk = 0; kk < TP; kk += 16) {
    int i = i0 + ty, ka = kk + tx;
    As[ty][tx] = (i < TP && ka < TP) ? A[i * TP + ka] : -1.0e30f;
    int kb = kk + ty, j = j0 + tx;
    Bs[ty][tx] = (kb < TP && j < TP) ? Bm[kb * TP + j] : -1.0e30f;
    __syncthreads();
#pragma unroll
    for (int k = 0; k < 16; ++k)
      acc = fmaxf(acc, As[ty][k] + Bs[k][tx]);
    __syncthreads();
  }

  int i = i0 + ty, j = j0 + tx;
  if (i < TP && j < TP) {
    int h = bh & 15, b = bh >> 4;
    float v = 1.0f / (1.0f + __expf(-(acc + bias[h])));
    OUT[((size_t)((b * TP + i) * TP) + j) * 16 + h] = v;
  }
}

// ---------------------------------------------------------------------------
extern "C" void kernel_launch(void* const* d_in, const int* in_sizes, int n_in,
                              void* d_out, int out_size, void* d_ws, size_t ws_size,
                              hipStream_t stream) {
  const float* a  = (const float*)d_in[0];
  const float* W0 = (const float*)d_in[1];
  const float* b0 = (const float*)d_in[2];
  const float* W1 = (const float*)d_in[3];
  const float* b1 = (const float*)d_in[4];
  const float* W2 = (const float*)d_in[5];
  const float* b2 = (const float*)d_in[6];
  float* out = (float*)d_out;

  // workspace layout (floats)
  float* rp = (float*)d_ws;                    // (4,97,97,32) = 1,204,352
  float* ft = rp + (size_t)NROWS * RELC;       // (64, 97,97)  =   602,176
  float* rt = ft + (size_t)16 * NB * MAT;      //   602,176
  float* f1 = rt + (size_t)16 * NB * MAT;      //   602,176  (f1 then f2)

  hipMemsetAsync(rp, 0, (size_t)NROWS * RELC * sizeof(float), stream);

  pool_pass1<<<24, 256, 0, stream>>>(a, rp);
  pool_pass2<<<24, 256, 0, stream>>>(rp);

  const int ntiles = (NROWS + 15) / 16;        // 2353
  const int gblocks = (ntiles + 3) / 4;        // 4 waves per block
  dim3 tgrid(7, 7, NB * 16);
  dim3 tblk(16, 16);

  // --- dp_step 0: ft = r @ W0[:32], rt = r @ W0[32:], tropical + sigmoid ---
  wmma_gemm16<<<gblocks, 128, 0, stream>>>(rp, 32, nullptr, 0, W0,            ft, 0, nullptr);
  wmma_gemm16<<<gblocks, 128, 0, stream>>>(rp, 32, nullptr, 0, W0 + 32 * 16,  rt, 0, nullptr);
  tropical_maxplus<<<tgrid, tblk, 0, stream>>>(ft, rt, b0, f1);

  // --- dp_step 1: ft = [f1|r] @ W1[:48], rt = r @ W1[48:] ---
  wmma_gemm16<<<gblocks, 128, 0, stream>>>(f1, 16, rp, 32, W1,                ft, 0, nullptr);
  wmma_gemm16<<<gblocks, 128, 0, stream>>>(rp, 32, nullptr, 0, W1 + 48 * 16,  rt, 0, nullptr);
  tropical_maxplus<<<tgrid, tblk, 0, stream>>>(ft, rt, b1, f1);   // f1 <- f2

  // --- output: sigmoid([f2|r] @ W2 + b2), sliced [:, :96, 1:, :] ---
  wmma_gemm16<<<gblocks, 128, 0, stream>>>(f1, 16, rp, 32, W2, out, 2, b2);
}